// GAT_3384434229767
// MI455X (gfx1250) — compile-verified
//
#include <hip/hip_runtime.h>
#include <hip/hip_bf16.h>

#define IN_FEATS   128
#define OUT_FEATS  16
#define NEG_SLOPE  0.2f

typedef float v2f __attribute__((ext_vector_type(2)));
typedef float v8f __attribute__((ext_vector_type(8)));

// ---------------------------------------------------------------------------
// K1: hW = h @ W^T  via native fp32 WMMA (V_WMMA_F32_16X16X4_F32).
// One wave -> one 16x16 tile (16 nodes x 16 out feats), K=128 in 32 steps.
// A: 16x4 fp32 fragment: lanes 0-15 hold rows, VGPR0/1 = K,K+1; lanes 16-31
//    hold the same rows at K+2,K+3 (ISA 7.12.2).
// B: 4x16 fragment, B[k][n] = W[n][k]: lane n&15 loads contiguous W[n][k..k+1].
// C/D: VGPR g -> row m0+g (lanes 0-15) / m0+8+g (lanes 16-31), col = lane&15.
// ---------------------------------------------------------------------------
__global__ void gat_project_wmma(const float* __restrict__ h,
                                 const float* __restrict__ W,
                                 float* __restrict__ hW, int nNodes) {
  int wave = (blockIdx.x * blockDim.x + threadIdx.x) >> 5;
  int lane = threadIdx.x & 31;
  int m0 = wave * 16;
  if (m0 >= nNodes) return;            // wave-uniform: EXEC stays all-1s
  int half = lane >> 4;                // 0: K,K+1   1: K+2,K+3
  int l16  = lane & 15;
  const float* arow = h + (size_t)(m0 + l16) * IN_FEATS;
  const float* brow = W + (size_t)l16 * IN_FEATS;
  v8f acc = {};
#pragma unroll
  for (int k = 0; k < IN_FEATS; k += 4) {
    int ka = k + half * 2;
    v2f a = *(const v2f*)(arow + ka);
    v2f b = *(const v2f*)(brow + ka);
    acc = __builtin_amdgcn_wmma_f32_16x16x4_f32(false, a, false, b,
                                                (short)0, acc, false, false);
  }
#pragma unroll
  for (int g = 0; g < 8; ++g) {
    int row = m0 + half * 8 + g;
    hW[(size_t)row * OUT_FEATS + l16] = acc[g];
  }
}

// ---------------------------------------------------------------------------
// K2: per-node: el/er dots, init emax=-inf, esum=0, zero out[] (harness
// poisons d_out; every launch must rebuild it deterministically).
// ---------------------------------------------------------------------------
__global__ void gat_node_prep(const float* __restrict__ hW,
                              const float* __restrict__ a_l,
                              const float* __restrict__ a_r,
                              float* __restrict__ el, float* __restrict__ er,
                              float* __restrict__ emax, float* __restrict__ esum,
                              float* __restrict__ out, int nNodes) {
  int n = blockIdx.x * blockDim.x + threadIdx.x;
  if (n >= nNodes) return;
  const float* row = hW + (size_t)n * OUT_FEATS;
  float sl = 0.f, sr = 0.f;
#pragma unroll
  for (int f = 0; f < OUT_FEATS; ++f) {
    float v = row[f];
    sl += v * a_l[f];
    sr += v * a_r[f];
    out[(size_t)n * OUT_FEATS + f] = 0.f;
  }
  el[n] = sl;
  er[n] = sr;
  emax[n] = __int_as_float(0xFF800000);   // -inf sentinel
  esum[n] = 0.f;
}

// Sign-split float atomic max: lowers to native integer global atomics.
// int-max for non-negative bit patterns, uint-min for negative ones; the
// -inf init (0xFF800000) is consistent under both orderings.
__device__ __forceinline__ void atomicMaxFloat(float* addr, float val) {
  if (val >= 0.f) atomicMax((int*)addr, __float_as_int(val));
  else            atomicMin((unsigned int*)addr, (unsigned int)__float_as_int(val));
}

// ---------------------------------------------------------------------------
// K3: per-edge logit + leaky relu + segment max (into emax[dst]).
// ---------------------------------------------------------------------------
__global__ void gat_edge_logit(const int* __restrict__ src,
                               const int* __restrict__ dst,
                               const float* __restrict__ el,
                               const float* __restrict__ er,
                               float* __restrict__ elog,
                               float* __restrict__ emax, int nEdges) {
  int e = blockIdx.x * blockDim.x + threadIdx.x;
  if (e >= nEdges) return;
  int s = src[e], d = dst[e];
  float x = el[s] + er[d];
  x = (x > 0.f) ? x : NEG_SLOPE * x;
  elog[e] = x;
  atomicMaxFloat(emax + d, x);
}

// K4: nodes with no incoming edges: emax := 0 (matches reference isfinite fix)
__global__ void gat_node_fixmax(float* __restrict__ emax, int nNodes) {
  int n = blockIdx.x * blockDim.x + threadIdx.x;
  if (n >= nNodes) return;
  if (emax[n] == __int_as_float(0xFF800000)) emax[n] = 0.f;
}

// K5: per-edge exp(e - emax[dst]) and segment sum into esum[dst].
__global__ void gat_edge_exp(const int* __restrict__ dst,
                             float* __restrict__ elog,
                             const float* __restrict__ emax,
                             float* __restrict__ esum, int nEdges) {
  int e = blockIdx.x * blockDim.x + threadIdx.x;
  if (e >= nEdges) return;
  int d = dst[e];
  float ex = __expf(elog[e] - emax[d]);
  elog[e] = ex;                 // overwrite logit with exp value
  atomicAdd(esum + d, ex);
}

// ---------------------------------------------------------------------------
// K6: weighted aggregation. 16 lanes cooperate on one edge so the hW[src]
// gather and out[dst] atomic adds are contiguous 64B transactions in L2.
// ---------------------------------------------------------------------------
__global__ void gat_edge_aggr(const int* __restrict__ src,
                              const int* __restrict__ dst,
                              const float* __restrict__ elog,
                              const float* __restrict__ esum,
                              const float* __restrict__ hW,
                              float* __restrict__ out, int nEdges) {
  long long t = (long long)blockIdx.x * blockDim.x + threadIdx.x;
  int e = (int)(t >> 4);
  int f = (int)(t & 15);
  if (e >= nEdges) return;
  int s = src[e], d = dst[e];
  float alpha = elog[e] / fmaxf(esum[d], 1e-16f);
  atomicAdd(out + (size_t)d * OUT_FEATS + f,
            hW[(size_t)s * OUT_FEATS + f] * alpha);
}

extern "C" void kernel_launch(void* const* d_in, const int* in_sizes, int n_in,
                              void* d_out, int out_size, void* d_ws, size_t ws_size,
                              hipStream_t stream) {
  const float* h   = (const float*)d_in[0];
  const int*   src = (const int*)d_in[1];
  const int*   dst = (const int*)d_in[2];
  const float* W   = (const float*)d_in[3];
  const float* a_l = (const float*)d_in[4];
  const float* a_r = (const float*)d_in[5];
  float* out = (float*)d_out;

  int nNodes = in_sizes[0] / IN_FEATS;
  int nEdges = in_sizes[1];

  // workspace layout (floats)
  float* ws   = (float*)d_ws;
  float* hW   = ws;                                   // nNodes * 16
  float* el   = hW + (size_t)nNodes * OUT_FEATS;      // nNodes
  float* er   = el + nNodes;                          // nNodes
  float* emax = er + nNodes;                          // nNodes
  float* esum = emax + nNodes;                        // nNodes
  float* elog = esum + nNodes;                        // nEdges

  dim3 blk(256);
  int tiles = (nNodes + 15) / 16;                     // 16 nodes per wave
  int blocksK1 = (tiles + 7) / 8;                     // 8 waves per block

  gat_project_wmma<<<blocksK1, blk, 0, stream>>>(h, W, hW, nNodes);
  gat_node_prep<<<(nNodes + 255) / 256, blk, 0, stream>>>(hW, a_l, a_r, el, er,
                                                          emax, esum, out, nNodes);
  gat_edge_logit<<<(nEdges + 255) / 256, blk, 0, stream>>>(src, dst, el, er,
                                                           elog, emax, nEdges);
  gat_node_fixmax<<<(nNodes + 255) / 256, blk, 0, stream>>>(emax, nNodes);
  gat_edge_exp<<<(nEdges + 255) / 256, blk, 0, stream>>>(dst, elog, emax, esum, nEdges);

  long long laneWork = (long long)nEdges * OUT_FEATS;
  int blocksK6 = (int)((laneWork + 255) / 256);
  gat_edge_aggr<<<blocksK6, blk, 0, stream>>>(src, dst, elog, esum, hW, out, nEdges);
}